// GATCell_11364483465749
// MI455X (gfx1250) — compile-verified
//
#include <hip/hip_runtime.h>
#include <hip/hip_bf16.h>
#include <math.h>

// ---------------------------------------------------------------------------
// GAT-GRU cell for MI455X (gfx1250).
// All four matmuls run on V_WMMA_F32_16X16X4_F32 (native f32 matrix path,
// keeps reference f32 numerics). Softmax is two-pass (row stats + streaming
// aggregation) so the [16,1024,1024] attention tensor is never materialized.
// Aggregation stages h j-tiles into LDS once per block (8 waves share them),
// double-buffered via GLOBAL_LOAD_ASYNC_TO_LDS_B128 + s_wait_asynccnt.
// ---------------------------------------------------------------------------

#define Bq   16
#define Nn   1024
#define Uu   64
#define ALPHA 0.2f
#define NEGV (-9.0e15f)

typedef __attribute__((ext_vector_type(2))) float v2f;
typedef __attribute__((ext_vector_type(8))) float v8f;
typedef __attribute__((ext_vector_type(4))) int   v4i;

// async-copy builtin wants typed AS pointers: global int4* src, LDS int4* dst
typedef __attribute__((address_space(1))) v4i gv4i;   // global
typedef __attribute__((address_space(3))) v4i sv4i;   // LDS

#if defined(__AMDGCN__) && __has_builtin(__builtin_amdgcn_global_load_async_to_lds_b128)
#define HAVE_ASYNC_COPY 1
#else
#define HAVE_ASYNC_COPY 0
#endif

static __device__ __forceinline__ float lrelu(float x) {
    return x > 0.0f ? x : ALPHA * x;
}

// D = A(16x4 f32) * B(4x16 f32) + C(16x16 f32)
static __device__ __forceinline__ v8f wmma_f32x4(v2f a, v2f b, v8f c) {
#if defined(__gfx1250__)
    // 8 args: (neg_a, A, neg_b, B, c_mod, C, reuse_a, reuse_b)
    return __builtin_amdgcn_wmma_f32_16x16x4_f32(false, a, false, b,
                                                 (short)0, c, false, false);
#else
    return c; // host pass stub (never executed on device)
#endif
}

static __device__ __forceinline__ void wait_async0() {
#if defined(__gfx1250__)
#if __has_builtin(__builtin_amdgcn_s_wait_asynccnt)
    __builtin_amdgcn_s_wait_asynccnt(0);
#else
    asm volatile("s_wait_asynccnt 0x0" ::: "memory");
#endif
#endif
}

// Stage one 16-row x F-col tile of H (contiguous 16*F floats) into LDS.
// 256 threads cooperate; async path uses B128 LDS-DMA tracked by ASYNCcnt.
template <int NT>
static __device__ __forceinline__ void stage_tile(const float* __restrict__ Hb, int j0,
                                                  float* dst, int tid) {
    const int F = NT * 16;
    const float* src = Hb + (size_t)j0 * F;
    for (int c = tid; c < (16 * F) / 4; c += 256) {
        const int off = c * 4;
#if HAVE_ASYNC_COPY
        __builtin_amdgcn_global_load_async_to_lds_b128(
            (gv4i*)(src + off), (sv4i*)(dst + off), 0, 0);
#else
        *(float4*)(dst + off) = *(const float4*)(src + off);
#endif
    }
}

// ---------------------------------------------------------------------------
// Kernel 1: h[b,n,o] = sum_f concat(X,Y)[b,n,f] * W[f,o]   (f in 0..127)
// grid: B*(N/128) blocks x 256 threads (8 waves). Each wave owns a 16-row
// block and sweeps all F/16 column tiles; W is staged in dynamic LDS.
// ---------------------------------------------------------------------------
__global__ void gemm_concat(const float* __restrict__ X, const float* __restrict__ Y,
                            const float* __restrict__ W, float* __restrict__ H,
                            int F) {
    extern __shared__ float Wlds[];                    // 128 * F floats
    const int tid   = threadIdx.x;
    const int lane  = tid & 31;
    const int wave  = tid >> 5;
    const int b     = blockIdx.x / (Nn / 128);
    const int i0    = (blockIdx.x % (Nn / 128)) * 128 + wave * 16;

    for (int idx = tid; idx < 128 * F; idx += blockDim.x) Wlds[idx] = W[idx];
    __syncthreads();

    const int row   = lane & 15;        // A: M index / B,C: N index
    const int khalf = (lane >> 4) * 2;  // 0 or 2
    const float* xrow = X + ((size_t)b * Nn + i0 + row) * Uu;
    const float* yrow = Y + ((size_t)b * Nn + i0 + row) * Uu;

    for (int t = 0; t < F / 16; ++t) {
        const int o0 = t * 16;
        v8f acc = {};
        for (int ks = 0; ks < 32; ++ks) {              // K = 128, 4 per WMMA
            const int k0 = ks * 4;
            const int ka = k0 + khalf;                 // even; never straddles U
            v2f a, bm;
            const float* src = (ka < Uu) ? (xrow + ka) : (yrow + ka - Uu);
            a.x  = src[0];
            a.y  = src[1];
            bm.x = Wlds[(size_t)(ka + 0) * F + o0 + row];
            bm.y = Wlds[(size_t)(ka + 1) * F + o0 + row];
            acc = wmma_f32x4(a, bm, acc);
        }
        float* hout = H + ((size_t)b * Nn + i0) * F + o0;
        const int m0 = (lane >> 4) * 8;                // C rows: v + 8*(lane>=16)
#pragma unroll
        for (int v = 0; v < 8; ++v)
            hout[(size_t)(m0 + v) * F + row] = acc[v];
    }
}

// ---------------------------------------------------------------------------
// Kernel 2: s_src[b,n] = h . a[0:F], s_dst[b,n] = h . a[F:2F]
// One wave per node, shuffle reduction.
// ---------------------------------------------------------------------------
__global__ void scores(const float* __restrict__ H, const float* __restrict__ a,
                       float* __restrict__ ssrc, float* __restrict__ sdst, int F) {
    const int wg = blockIdx.x * (blockDim.x >> 5) + (threadIdx.x >> 5);
    if (wg >= Bq * Nn) return;
    const int lane = threadIdx.x & 31;
    const float* h = H + (size_t)wg * F;
    float ss = 0.0f, sd = 0.0f;
    for (int f = lane; f < F; f += 32) {
        const float hv = h[f];
        ss += hv * a[f];
        sd += hv * a[F + f];
    }
    for (int m = 16; m >= 1; m >>= 1) {
        ss += __shfl_xor(ss, m, 32);
        sd += __shfl_xor(sd, m, 32);
    }
    if (lane == 0) { ssrc[wg] = ss; sdst[wg] = sd; }
}

// ---------------------------------------------------------------------------
// Kernel 3: masked-softmax row statistics. One wave per (b,i) row:
// m = max_j e_ij, l = sum_j exp(e_ij - m), with e masked by adjacency.
// ---------------------------------------------------------------------------
__global__ void rowstats(const float* __restrict__ ssrc, const float* __restrict__ sdst,
                         const float* __restrict__ adj,
                         float* __restrict__ mrow, float* __restrict__ lrow) {
    const int wg = blockIdx.x * (blockDim.x >> 5) + (threadIdx.x >> 5);
    if (wg >= Bq * Nn) return;
    const int lane = threadIdx.x & 31;
    const int b = wg / Nn, i = wg % Nn;
    const float si = ssrc[wg];
    const float* sd   = sdst + (size_t)b * Nn;
    const float* arow = adj + (size_t)i * Nn;
    float m = -3.0e38f, l = 0.0f;
    for (int j = lane; j < Nn; j += 32) {
        const float e = (arow[j] > 0.0f) ? lrelu(si + sd[j]) : NEGV;
        if (e > m) { l *= __expf(m - e); m = e; }
        l += __expf(e - m);
    }
    for (int d = 16; d >= 1; d >>= 1) {
        const float mo = __shfl_xor(m, d, 32);
        const float lo = __shfl_xor(l, d, 32);
        const float mn = fmaxf(m, mo);
        l = l * __expf(m - mn) + lo * __expf(mo - mn);
        m = mn;
    }
    if (lane == 0) { mrow[wg] = m; lrow[wg] = l; }
}

// ---------------------------------------------------------------------------
// Kernel 4: out[b,i,:] = sum_j softmax(e)_ij * h[b,j,:]  via WMMA.
// One block = 8 waves = 128 consecutive rows of one batch. The 16xF h j-tile
// is staged once per block into LDS (double-buffered, async LDS-DMA) and
// shared by all 8 waves; attention probabilities are generated on the fly in
// the WMMA A-fragment layout.
// ---------------------------------------------------------------------------
template <int NT>   // NT = F/16 column tiles (8 for F=128, 4 for F=64)
__global__ void __launch_bounds__(256)
aggregate(const float* __restrict__ H, const float* __restrict__ ssrc,
          const float* __restrict__ sdst, const float* __restrict__ adj,
          const float* __restrict__ mrow, const float* __restrict__ lrow,
          float* __restrict__ Out) {
    const int F = NT * 16;
    __shared__ float tile[2][16 * NT * 16];            // two 16xF buffers

    const int tid   = threadIdx.x;
    const int lane  = tid & 31;
    const int wave  = tid >> 5;
    const int b     = blockIdx.x / (Nn / 128);         // all waves share batch b
    const int i0    = (blockIdx.x % (Nn / 128)) * 128 + wave * 16;
    const int row   = lane & 15;
    const int khalf = (lane >> 4) * 2;

    const float si    = ssrc[(size_t)b * Nn + i0 + row];
    const float mi    = mrow[(size_t)b * Nn + i0 + row];
    const float inv_l = 1.0f / lrow[(size_t)b * Nn + i0 + row];
    const float* sd   = sdst + (size_t)b * Nn;
    const float* arow = adj + (size_t)(i0 + row) * Nn;
    const float* Hb   = H + (size_t)b * Nn * F;

    stage_tile<NT>(Hb, 0, &tile[0][0], tid);           // prologue fill
    int cur = 0;

    v8f acc[NT] = {};
    for (int j0 = 0; j0 < Nn; j0 += 16) {
        wait_async0();                                 // tile[cur] resident
        __syncthreads();
        if (j0 + 16 < Nn)                              // prefetch next tile
            stage_tile<NT>(Hb, j0 + 16, &tile[cur ^ 1][0], tid);

        v2f af[4];
#pragma unroll
        for (int ks = 0; ks < 4; ++ks) {               // probability A-fragments
            const int j = j0 + ks * 4 + khalf;
            af[ks].x = (arow[j + 0] > 0.0f)
                         ? __expf(lrelu(si + sd[j + 0]) - mi) * inv_l : 0.0f;
            af[ks].y = (arow[j + 1] > 0.0f)
                         ? __expf(lrelu(si + sd[j + 1]) - mi) * inv_l : 0.0f;
        }

        const float* tl = &tile[cur][0];
#pragma unroll
        for (int t = 0; t < NT; ++t) {
#pragma unroll
            for (int ks = 0; ks < 4; ++ks) {
                const int kr = ks * 4 + khalf;         // K row within tile
                v2f bf;
                bf.x = tl[(kr + 0) * F + t * 16 + row];
                bf.y = tl[(kr + 1) * F + t * 16 + row];
                acc[t] = wmma_f32x4(af[ks], bf, acc[t]);
            }
        }
        __syncthreads();                               // everyone done with cur
        cur ^= 1;
    }

    const int m0 = (lane >> 4) * 8;
    float* ob = Out + ((size_t)b * Nn + i0) * F;
#pragma unroll
    for (int t = 0; t < NT; ++t)
#pragma unroll
        for (int v = 0; v < 8; ++v)
            ob[(size_t)(m0 + v) * F + t * 16 + row] = acc[t][v];
}

// ---------------------------------------------------------------------------
// Kernel 5: gv = sigmoid(out1); r,z = split; rs = r*state; keep z.
// ---------------------------------------------------------------------------
__global__ void gate_rz(const float* __restrict__ Out1, const float* __restrict__ state,
                        float* __restrict__ rs, float* __restrict__ zbuf) {
    const size_t idx = (size_t)blockIdx.x * blockDim.x + threadIdx.x;
    if (idx >= (size_t)Bq * Nn * Uu) return;
    const size_t node = idx / Uu;
    const int u = (int)(idx % Uu);
    const float* o = Out1 + node * (2 * Uu);
    const float r = 1.0f / (1.0f + __expf(-o[u]));
    const float z = 1.0f / (1.0f + __expf(-o[Uu + u]));
    rs[idx]   = r * state[idx];
    zbuf[idx] = z;
}

// ---------------------------------------------------------------------------
// Kernel 6: new_h = z*state + (1-z)*tanh(out2)
// ---------------------------------------------------------------------------
__global__ void final_combine(const float* __restrict__ zbuf, const float* __restrict__ state,
                              const float* __restrict__ Out2, float* __restrict__ out) {
    const size_t idx = (size_t)blockIdx.x * blockDim.x + threadIdx.x;
    if (idx >= (size_t)Bq * Nn * Uu) return;
    const float z = zbuf[idx];
    const float h = tanhf(Out2[idx]);
    out[idx] = z * state[idx] + (1.0f - z) * h;
}

// ---------------------------------------------------------------------------
extern "C" void kernel_launch(void* const* d_in, const int* in_sizes, int n_in,
                              void* d_out, int out_size, void* d_ws, size_t ws_size,
                              hipStream_t stream) {
    const float* X     = (const float*)d_in[0];
    const float* state = (const float*)d_in[1];
    const float* adj   = (const float*)d_in[2];
    const float* W1    = (const float*)d_in[3];
    const float* a1    = (const float*)d_in[4];
    const float* W2    = (const float*)d_in[5];
    const float* a2    = (const float*)d_in[6];
    float* out = (float*)d_out;

    // workspace partition (floats)
    float* ws = (float*)d_ws;
    size_t off = 0;
    float* h1    = ws + off; off += (size_t)Bq * Nn * 128;
    float* out1  = ws + off; off += (size_t)Bq * Nn * 128;
    float* h2    = ws + off; off += (size_t)Bq * Nn * 64;
    float* out2  = ws + off; off += (size_t)Bq * Nn * 64;
    float* rs    = ws + off; off += (size_t)Bq * Nn * 64;
    float* zbuf  = ws + off; off += (size_t)Bq * Nn * 64;
    float* ssrc1 = ws + off; off += (size_t)Bq * Nn;
    float* sdst1 = ws + off; off += (size_t)Bq * Nn;
    float* m1    = ws + off; off += (size_t)Bq * Nn;
    float* l1    = ws + off; off += (size_t)Bq * Nn;
    float* ssrc2 = ws + off; off += (size_t)Bq * Nn;
    float* sdst2 = ws + off; off += (size_t)Bq * Nn;
    float* m2    = ws + off; off += (size_t)Bq * Nn;
    float* l2    = ws + off; off += (size_t)Bq * Nn;

    const int gemm_grid  = Bq * (Nn / 128);            // 128 blocks
    const int wave_grid  = (Bq * Nn) / 8;              // 8 waves/block
    const int agg_grid   = Bq * (Nn / 128);            // 128 blocks, 8 waves each
    const int elem_grid  = (Bq * Nn * Uu + 255) / 256;

    // ---- Layer 1 (F = 128) ----
    gemm_concat<<<gemm_grid, 256, 128 * 128 * sizeof(float), stream>>>(X, state, W1, h1, 128);
    scores<<<wave_grid, 256, 0, stream>>>(h1, a1, ssrc1, sdst1, 128);
    rowstats<<<wave_grid, 256, 0, stream>>>(ssrc1, sdst1, adj, m1, l1);
    aggregate<8><<<agg_grid, 256, 0, stream>>>(h1, ssrc1, sdst1, adj, m1, l1, out1);
    gate_rz<<<elem_grid, 256, 0, stream>>>(out1, state, rs, zbuf);

    // ---- Layer 2 (F = 64) ----
    gemm_concat<<<gemm_grid, 256, 128 * 64 * sizeof(float), stream>>>(X, rs, W2, h2, 64);
    scores<<<wave_grid, 256, 0, stream>>>(h2, a2, ssrc2, sdst2, 64);
    rowstats<<<wave_grid, 256, 0, stream>>>(ssrc2, sdst2, adj, m2, l2);
    aggregate<4><<<agg_grid, 256, 0, stream>>>(h2, ssrc2, sdst2, adj, m2, l2, out2);
    final_combine<<<elem_grid, 256, 0, stream>>>(zbuf, state, out2, out);
}